// AegisGuardGNN_49486613185072
// MI455X (gfx1250) — compile-verified
//
#include <hip/hip_runtime.h>
#include <math.h>

// ---------------------------------------------------------------------------
// CDNA5 (gfx1250) GNN forward: encoder -> GAT(4 heads) -> 2x SAGE -> pool -> MLP
// Dense [*,128]@[128,128] GEMMs use V_WMMA_F32_16X16X4_F32 (exact f32 math).
// ---------------------------------------------------------------------------

typedef __attribute__((ext_vector_type(2))) float v2f;
typedef __attribute__((ext_vector_type(8))) float v8f;

#define LDS_STRIDE 132            // 132 % 64 banks = 4 -> conflict-free column reads
#define FENC_NEG_INF 0x007FFFFFu  // order-preserving encoding of -inf

__device__ __forceinline__ unsigned fenc(float f) {
    unsigned u = __float_as_uint(f);
    return (u & 0x80000000u) ? ~u : (u | 0x80000000u);
}
__device__ __forceinline__ float fdec(unsigned e) {
    unsigned u = (e & 0x80000000u) ? (e ^ 0x80000000u) : ~e;
    return __uint_as_float(u);
}

// ---------------------------------------------------------------------------
// WMMA GEMM: out[16 rows x 128 cols per block] =
//   epilogue( A1@B1 (+ A2@B2) + bias ), epilogue = optional BN(inference)+ReLU.
// 256 threads = 8 waves; each wave owns one 16x16 tile, K=128 in steps of 4.
// A1 rows optionally scaled by 1/max(rowdiv,1) (SAGE mean aggregation).
// ---------------------------------------------------------------------------
template <bool DUAL>
__global__ __launch_bounds__(256)
void gemm_n128_wmma(const float* __restrict__ A1, const float* __restrict__ B1,
                    const float* __restrict__ A2, const float* __restrict__ B2,
                    const float* __restrict__ rowdiv,
                    const float* __restrict__ bias,
                    const float* __restrict__ bng, const float* __restrict__ bnb,
                    const float* __restrict__ bnm, const float* __restrict__ bnv,
                    float* __restrict__ out, int nrows, int relu)
{
    __shared__ float lA1[16 * LDS_STRIDE];
    __shared__ float lA2[DUAL ? 16 * LDS_STRIDE : 4];

    const int row0 = blockIdx.x * 16;
    const int t = threadIdx.x;

    // Stage A tile(s): each thread copies 8 contiguous floats (fully coalesced:
    // a 16x128 row tile is one contiguous 8KB block).
    {
        int r  = t >> 4;
        int c0 = (t & 15) << 3;
        int rg = min(row0 + r, nrows - 1);
        float inv = 1.0f;
        if (rowdiv) inv = 1.0f / fmaxf(rowdiv[rg], 1.0f);
        const float* s1 = A1 + (size_t)rg * 128 + c0;
        #pragma unroll
        for (int j = 0; j < 8; ++j) lA1[r * LDS_STRIDE + c0 + j] = s1[j] * inv;
        if (DUAL) {
            const float* s2 = A2 + (size_t)rg * 128 + c0;
            #pragma unroll
            for (int j = 0; j < 8; ++j) lA2[r * LDS_STRIDE + c0 + j] = s2[j];
        }
    }
    __syncthreads();

    const int lane = t & 31;
    const int n    = ((t >> 5) << 4) + (lane & 15); // output column
    const int kg   = lane >> 4;                     // 0 or 1
    const int m    = lane & 15;                     // A row within tile

    v8f acc = (v8f){0.f, 0.f, 0.f, 0.f, 0.f, 0.f, 0.f, 0.f};
    for (int k0 = 0; k0 < 128; k0 += 4) {
        const int k = k0 + (kg << 1);
        v2f a, b;
        a.x = lA1[m * LDS_STRIDE + k];
        a.y = lA1[m * LDS_STRIDE + k + 1];
        b.x = B1[k * 128 + n];
        b.y = B1[(k + 1) * 128 + n];
        acc = __builtin_amdgcn_wmma_f32_16x16x4_f32(false, a, false, b,
                                                    (short)0, acc, false, false);
        if (DUAL) {
            v2f a2, b2;
            a2.x = lA2[m * LDS_STRIDE + k];
            a2.y = lA2[m * LDS_STRIDE + k + 1];
            b2.x = B2[k * 128 + n];
            b2.y = B2[(k + 1) * 128 + n];
            acc = __builtin_amdgcn_wmma_f32_16x16x4_f32(false, a2, false, b2,
                                                        (short)0, acc, false, false);
        }
    }

    float add = bias ? bias[n] : 0.0f;
    float scale = 1.0f, shift = 0.0f;
    if (bng) {
        float s = bng[n] * rsqrtf(bnv[n] + 1e-5f);
        scale = s;
        shift = bnb[n] - bnm[n] * s;
    }
    #pragma unroll
    for (int v = 0; v < 8; ++v) {
        int mr = row0 + v + (kg << 3);
        float val = (acc[v] + add) * scale + shift;
        if (relu) val = fmaxf(val, 0.0f);
        if (mr < nrows) out[(size_t)mr * 128 + n] = val;
    }
}

// ---------------------------------------------------------------------------
// Small helpers / elementwise kernels
// ---------------------------------------------------------------------------
__global__ void fill_f32(float* p, float v, size_t n) {
    size_t i = (size_t)blockIdx.x * blockDim.x + threadIdx.x;
    if (i < n) p[i] = v;
}
__global__ void fill_u32(unsigned* p, unsigned v, size_t n) {
    size_t i = (size_t)blockIdx.x * blockDim.x + threadIdx.x;
    if (i < n) p[i] = v;
}

// a_src[n,h] = <hh[n,h,:], att_src[h,:]> ; a_dst likewise
__global__ void gat_attn_dots(const float* __restrict__ hh,
                              const float* __restrict__ att_src,
                              const float* __restrict__ att_dst,
                              float* __restrict__ asrc, float* __restrict__ adst, int N)
{
    int idx = blockIdx.x * blockDim.x + threadIdx.x;
    if (idx >= N * 4) return;
    int n = idx >> 2, h = idx & 3;
    const float* row = hh + (size_t)n * 128 + h * 32;
    const float* as = att_src + h * 32;
    const float* ad = att_dst + h * 32;
    float s = 0.f, d = 0.f;
    #pragma unroll
    for (int c = 0; c < 32; ++c) { float v = row[c]; s += v * as[c]; d += v * ad[c]; }
    asrc[idx] = s; adst[idx] = d;
}

__device__ __forceinline__ void edge_ends(const int* ei, int E, int e, int& s, int& d) {
    if (e < E) { s = ei[e]; d = ei[E + e]; }
    else       { s = d = e - E; }   // appended self loops
}

__global__ void gat_edge_max(const int* __restrict__ ei, int E, int N,
                             const float* __restrict__ asrc, const float* __restrict__ adst,
                             unsigned* __restrict__ nmax)
{
    int e = blockIdx.x * blockDim.x + threadIdx.x;
    if (e >= E + N) return;
    int s, d; edge_ends(ei, E, e, s, d);
    #pragma unroll
    for (int h = 0; h < 4; ++h) {
        float v = asrc[s * 4 + h] + adst[d * 4 + h];
        v = (v > 0.f) ? v : 0.2f * v;                   // leaky_relu(0.2)
        atomicMax(&nmax[d * 4 + h], fenc(v));
    }
}

__global__ void gat_edge_exp(const int* __restrict__ ei, int E, int N,
                             const float* __restrict__ asrc, const float* __restrict__ adst,
                             const unsigned* __restrict__ nmax,
                             float* __restrict__ eexp, float* __restrict__ nsum)
{
    int e = blockIdx.x * blockDim.x + threadIdx.x;
    if (e >= E + N) return;
    int s, d; edge_ends(ei, E, e, s, d);
    #pragma unroll
    for (int h = 0; h < 4; ++h) {
        float v = asrc[s * 4 + h] + adst[d * 4 + h];
        v = (v > 0.f) ? v : 0.2f * v;
        float w = __expf(v - fdec(nmax[d * 4 + h]));
        eexp[(size_t)e * 4 + h] = w;
        atomicAdd(&nsum[d * 4 + h], w);
    }
}

// one thread per (edge, channel)
__global__ void gat_edge_msg(const int* __restrict__ ei, int E, int N,
                             const float* __restrict__ hh,
                             const float* __restrict__ eexp, const float* __restrict__ nsum,
                             float* __restrict__ h2)
{
    size_t idx = (size_t)blockIdx.x * blockDim.x + threadIdx.x;
    size_t total = (size_t)(E + N) * 128;
    if (idx >= total) return;
    int e = (int)(idx >> 7), c = (int)(idx & 127), h = c >> 5;
    int s, d; edge_ends(ei, E, e, s, d);
    float w = eexp[(size_t)e * 4 + h] / (nsum[d * 4 + h] + 1e-16f);
    atomicAdd(&h2[(size_t)d * 128 + c], hh[(size_t)s * 128 + c] * w);
}

__global__ void bias_bn_relu(const float* __restrict__ in, const float* __restrict__ bias,
                             const float* __restrict__ g, const float* __restrict__ b,
                             const float* __restrict__ m, const float* __restrict__ v,
                             float* __restrict__ out, size_t total)
{
    size_t idx = (size_t)blockIdx.x * blockDim.x + threadIdx.x;
    if (idx >= total) return;
    int c = (int)(idx & 127);
    float x = in[idx] + (bias ? bias[c] : 0.f);
    float s = g[c] * rsqrtf(v[c] + 1e-5f);
    x = (x - m[c]) * s + b[c];
    out[idx] = fmaxf(x, 0.f);
}

__global__ void compute_deg(const int* __restrict__ ei, int E, float* __restrict__ deg) {
    int e = blockIdx.x * blockDim.x + threadIdx.x;
    if (e < E) atomicAdd(&deg[ei[E + e]], 1.0f);
}

__global__ void sage_agg(const int* __restrict__ ei, int E,
                         const float* __restrict__ h, float* __restrict__ agg)
{
    size_t idx = (size_t)blockIdx.x * blockDim.x + threadIdx.x;
    size_t total = (size_t)E * 128;
    if (idx >= total) return;
    int e = (int)(idx >> 7), c = (int)(idx & 127);
    atomicAdd(&agg[(size_t)ei[E + e] * 128 + c], h[(size_t)ei[e] * 128 + c]);
}

__global__ void pool_cnt(const int* __restrict__ batch, int N, float* __restrict__ pcnt) {
    int n = blockIdx.x * blockDim.x + threadIdx.x;
    if (n < N) atomicAdd(&pcnt[batch[n]], 1.0f);
}

__global__ void pool_sum_max(const int* __restrict__ batch, const float* __restrict__ h, int N,
                             float* __restrict__ psum, unsigned* __restrict__ pmax)
{
    size_t idx = (size_t)blockIdx.x * blockDim.x + threadIdx.x;
    if (idx >= (size_t)N * 128) return;
    int n = (int)(idx >> 7), c = (int)(idx & 127);
    int g = batch[n];
    float v = h[idx];
    atomicAdd(&psum[(size_t)g * 128 + c], v);
    atomicMax(&pmax[(size_t)g * 128 + c], fenc(v));
}

__global__ void pool_finalize(const float* __restrict__ psum, const unsigned* __restrict__ pmax,
                              const float* __restrict__ pcnt, float* __restrict__ ge, int G)
{
    int idx = blockIdx.x * blockDim.x + threadIdx.x;
    if (idx >= G * 256) return;
    int g = idx >> 8, c = idx & 255;
    float cnt = pcnt[g];
    float v;
    if (c < 128) v = psum[g * 128 + c] / fmaxf(cnt, 1.0f);
    else         v = (cnt > 0.f) ? fdec(pmax[g * 128 + (c - 128)]) : 0.0f;
    ge[(size_t)g * 256 + c] = v;
}

// tiny classifier layer: out[g,o] = act(in[g,:K] @ W[K,O] + b[o])
__global__ void dense_small(const float* __restrict__ in, const float* __restrict__ W,
                            const float* __restrict__ b, float* __restrict__ out,
                            int G, int K, int O, int relu)
{
    int idx = blockIdx.x * blockDim.x + threadIdx.x;
    if (idx >= G * O) return;
    int g = idx / O, o = idx % O;
    float acc = b[o];
    for (int k = 0; k < K; ++k) acc += in[(size_t)g * K + k] * W[(size_t)k * O + o];
    out[idx] = relu ? fmaxf(acc, 0.f) : acc;
}

// ---------------------------------------------------------------------------
static inline unsigned nblk(size_t n, int bs) { return (unsigned)((n + bs - 1) / bs); }

extern "C" void kernel_launch(void* const* d_in, const int* in_sizes, int n_in,
                              void* d_out, int out_size, void* d_ws, size_t ws_size,
                              hipStream_t stream)
{
    const float* x     = (const float*)d_in[0];
    const int*   ei    = (const int*)d_in[1];
    const int*   batch = (const int*)d_in[2];
    const int N = in_sizes[0] / 128;
    const int E = in_sizes[1] / 2;
    const int G = 128;

    // params in setup_inputs() dict order
    const float* encW  = (const float*)d_in[3];
    const float* encB  = (const float*)d_in[4];
    const float* ebn_g = (const float*)d_in[5];
    const float* ebn_b = (const float*)d_in[6];
    const float* ebn_m = (const float*)d_in[7];
    const float* ebn_v = (const float*)d_in[8];
    const float* gatW  = (const float*)d_in[9];
    const float* attS  = (const float*)d_in[10];
    const float* attD  = (const float*)d_in[11];
    const float* gatB  = (const float*)d_in[12];
    const float* bn0_g = (const float*)d_in[13];
    const float* bn0_b = (const float*)d_in[14];
    const float* bn0_m = (const float*)d_in[15];
    const float* bn0_v = (const float*)d_in[16];
    const float* Wl0 = (const float*)d_in[17];
    const float* bl0 = (const float*)d_in[18];
    const float* Wr0 = (const float*)d_in[19];
    const float* Wl1 = (const float*)d_in[20];
    const float* bl1 = (const float*)d_in[21];
    const float* Wr1 = (const float*)d_in[22];
    const float* bnA_g = (const float*)d_in[23];
    const float* bnA_b = (const float*)d_in[24];
    const float* bnA_m = (const float*)d_in[25];
    const float* bnA_v = (const float*)d_in[26];
    const float* bnB_g = (const float*)d_in[27];
    const float* bnB_b = (const float*)d_in[28];
    const float* bnB_m = (const float*)d_in[29];
    const float* bnB_v = (const float*)d_in[30];
    const float* W1 = (const float*)d_in[31];
    const float* b1 = (const float*)d_in[32];
    const float* W2 = (const float*)d_in[33];
    const float* b2 = (const float*)d_in[34];
    const float* W3 = (const float*)d_in[35];
    const float* b3 = (const float*)d_in[36];

    // ---- workspace layout (floats) ----
    float* w = (float*)d_ws;
    float* bufA = w;                 w += (size_t)N * 128;   // h1 / agg
    float* bufB = w;                 w += (size_t)N * 128;   // hh / h3
    float* bufC = w;                 w += (size_t)N * 128;   // h2 / h4
    float* asrc = w;                 w += (size_t)N * 4;
    float* adst = w;                 w += (size_t)N * 4;
    unsigned* nmax = (unsigned*)w;   w += (size_t)N * 4;
    float* nsum = w;                 w += (size_t)N * 4;
    float* eexp = w;                 w += (size_t)(E + N) * 4;
    float* deg  = w;                 w += (size_t)N;
    float* psum = w;                 w += (size_t)G * 128;
    unsigned* pmax = (unsigned*)w;   w += (size_t)G * 128;
    float* pcnt = w;                 w += (size_t)G;
    float* ge = w;                   w += (size_t)G * 256;
    float* z1 = w;                   w += (size_t)G * 128;
    float* z2 = w;                   w += (size_t)G * 64;

    const unsigned gemm_blocks = nblk(N, 16);
    const int EP = E + N;

    // 1) encoder: h1 = relu(bn(x@W + b))
    gemm_n128_wmma<false><<<gemm_blocks, 256, 0, stream>>>(
        x, encW, nullptr, nullptr, nullptr, encB,
        ebn_g, ebn_b, ebn_m, ebn_v, bufA, N, 1);

    // 2) GAT projection: hh = h1 @ Wg
    gemm_n128_wmma<false><<<gemm_blocks, 256, 0, stream>>>(
        bufA, gatW, nullptr, nullptr, nullptr, nullptr,
        nullptr, nullptr, nullptr, nullptr, bufB, N, 0);

    // 3) attention logits per node
    gat_attn_dots<<<nblk((size_t)N * 4, 256), 256, 0, stream>>>(bufB, attS, attD, asrc, adst, N);

    // 4) edge softmax + message aggregation
    fill_u32<<<nblk((size_t)N * 4, 256), 256, 0, stream>>>(nmax, FENC_NEG_INF, (size_t)N * 4);
    fill_f32<<<nblk((size_t)N * 4, 256), 256, 0, stream>>>(nsum, 0.f, (size_t)N * 4);
    fill_f32<<<nblk((size_t)N * 128, 256), 256, 0, stream>>>(bufC, 0.f, (size_t)N * 128);
    gat_edge_max<<<nblk(EP, 256), 256, 0, stream>>>(ei, E, N, asrc, adst, nmax);
    gat_edge_exp<<<nblk(EP, 256), 256, 0, stream>>>(ei, E, N, asrc, adst, nmax, eexp, nsum);
    gat_edge_msg<<<nblk((size_t)EP * 128, 256), 256, 0, stream>>>(ei, E, N, bufB, eexp, nsum, bufC);
    bias_bn_relu<<<nblk((size_t)N * 128, 256), 256, 0, stream>>>(
        bufC, gatB, bn0_g, bn0_b, bn0_m, bn0_v, bufC, (size_t)N * 128);

    // 5) degrees (original edges only)
    fill_f32<<<nblk(N, 256), 256, 0, stream>>>(deg, 0.f, (size_t)N);
    compute_deg<<<nblk(E, 256), 256, 0, stream>>>(ei, E, deg);

    // 6) SAGE layer 0: h3 = relu(bn((agg/deg)@Wl + h2@Wr + bl))
    fill_f32<<<nblk((size_t)N * 128, 256), 256, 0, stream>>>(bufA, 0.f, (size_t)N * 128);
    sage_agg<<<nblk((size_t)E * 128, 256), 256, 0, stream>>>(ei, E, bufC, bufA);
    gemm_n128_wmma<true><<<gemm_blocks, 256, 0, stream>>>(
        bufA, Wl0, bufC, Wr0, deg, bl0, bnA_g, bnA_b, bnA_m, bnA_v, bufB, N, 1);

    // 7) SAGE layer 1
    fill_f32<<<nblk((size_t)N * 128, 256), 256, 0, stream>>>(bufA, 0.f, (size_t)N * 128);
    sage_agg<<<nblk((size_t)E * 128, 256), 256, 0, stream>>>(ei, E, bufB, bufA);
    gemm_n128_wmma<true><<<gemm_blocks, 256, 0, stream>>>(
        bufA, Wl1, bufB, Wr1, deg, bl1, bnB_g, bnB_b, bnB_m, bnB_v, bufC, N, 1);

    // 8) per-graph mean/max pooling
    fill_f32<<<nblk((size_t)G * 128, 256), 256, 0, stream>>>(psum, 0.f, (size_t)G * 128);
    fill_u32<<<nblk((size_t)G * 128, 256), 256, 0, stream>>>(pmax, FENC_NEG_INF, (size_t)G * 128);
    fill_f32<<<nblk(G, 128), 128, 0, stream>>>(pcnt, 0.f, (size_t)G);
    pool_cnt<<<nblk(N, 256), 256, 0, stream>>>(batch, N, pcnt);
    pool_sum_max<<<nblk((size_t)N * 128, 256), 256, 0, stream>>>(batch, bufC, N, psum, pmax);
    pool_finalize<<<nblk((size_t)G * 256, 256), 256, 0, stream>>>(psum, pmax, pcnt, ge, G);

    // 9) classifier MLP
    dense_small<<<nblk((size_t)G * 128, 128), 128, 0, stream>>>(ge, W1, b1, z1, G, 256, 128, 1);
    dense_small<<<nblk((size_t)G * 64, 128), 128, 0, stream>>>(z1, W2, b2, z2, G, 128, 64, 1);
    dense_small<<<nblk((size_t)G * 2, 64), 64, 0, stream>>>(z2, W3, b3, (float*)d_out, G, 64, 2, 0);

    (void)n_in; (void)out_size; (void)ws_size;
}